// MPCSTGCNCNN_24472723652662
// MI455X (gfx1250) — compile-verified
//
#include <hip/hip_runtime.h>
#include <hip/hip_bf16.h>

// MPCSTGCNCNN for MI455X (gfx1250, wave32).
// Heavy parts: (a) conv over MPG (384 MB -> fused mean/max reduction, LDS row
// staging), (b) S-GEMMs done with V_WMMA_F32_16X16X4_F32.

typedef float v2f __attribute__((ext_vector_type(2)));
typedef float v8f __attribute__((ext_vector_type(8)));

#define NN 500
#define BB 32
#define TT 12
#define EMB 10

// ---------------------------------------------------------------------------
// Kernel 1: per-row softmax (or adj*exp path), writes S and rowsum[i].
// ---------------------------------------------------------------------------
__global__ __launch_bounds__(512)
void k_rowS(const float* __restrict__ E, const float* __restrict__ adj,
            float* __restrict__ S, float* __restrict__ rowsum,
            const int* __restrict__ fixedp, const int* __restrict__ stayp) {
    int i = blockIdx.x;
    int j = threadIdx.x;           // 0..511
    __shared__ float red[512];
    __shared__ float ei[EMB];
    if (j < EMB) ei[j] = E[i * EMB + j];
    __syncthreads();

    bool valid = j < NN;
    float s0 = 0.f;
    if (valid) {
        float d = 0.f;
        for (int dd = 0; dd < EMB; ++dd) d += ei[dd] * E[j * EMB + dd];
        s0 = fmaxf(d, 0.f);
        if (j == i) s0 = (float)stayp[0];
    }

    float val;
    if (fixedp[0] == 1) {
        val = valid ? adj[i * NN + j] * expf(s0) : 0.f;
    } else {
        red[j] = valid ? s0 : -1e30f;
        __syncthreads();
        for (int s = 256; s > 0; s >>= 1) {
            if (j < s) red[j] = fmaxf(red[j], red[j + s]);
            __syncthreads();
        }
        float m = red[0]; __syncthreads();
        float e = valid ? expf(s0 - m) : 0.f;
        red[j] = e; __syncthreads();
        for (int s = 256; s > 0; s >>= 1) {
            if (j < s) red[j] += red[j + s];
            __syncthreads();
        }
        float sum = red[0]; __syncthreads();
        val = e / sum;
    }

    red[j] = valid ? val : 0.f;
    __syncthreads();
    for (int s = 256; s > 0; s >>= 1) {
        if (j < s) red[j] += red[j + s];
        __syncthreads();
    }
    if (j == 0) rowsum[i] = red[0];
    if (valid) S[i * NN + j] = val;
}

// S[i,j] /= rowsum[j]   (replicates reference broadcasting: divide by COLUMN j's row-sum)
__global__ void k_coldiv(float* __restrict__ S, const float* __restrict__ rowsum) {
    int idx = blockIdx.x * blockDim.x + threadIdx.x;
    if (idx < NN * NN) S[idx] /= rowsum[idx % NN];
}

// ---------------------------------------------------------------------------
// Gather: Xt[m][b] = x[b][m];  xt[b][n] = sum_t Tp[t]*xw[b,t,n];
//         XW[(b*3+f)][m] = x_window[b, bidx[f], m]
// ---------------------------------------------------------------------------
__global__ void k_gather(const float* __restrict__ x, const float* __restrict__ xwin,
                         const float* __restrict__ Tp, const int* __restrict__ bidx,
                         float* __restrict__ Xt, float* __restrict__ XW,
                         float* __restrict__ xt) {
    int tid = blockIdx.x * blockDim.x + threadIdx.x;
    if (tid >= BB * NN) return;
    int b = tid / NN, m = tid % NN;
    Xt[m * BB + b] = x[b * NN + m];
    float s = 0.f;
    for (int t = 0; t < TT; ++t) s += Tp[t] * xwin[(b * TT + t) * NN + m];
    xt[b * NN + m] = s;
    for (int f = 0; f < 3; ++f)
        XW[(b * 3 + f) * NN + m] = xwin[(b * TT + bidx[f]) * NN + m];
}

// ---------------------------------------------------------------------------
// WMMA f32 GEMM: C(MxN) = A(MxK)*B(KxN), row-major. One wave per 16x16 tile.
// K must be a multiple of 4 (all calls use K=500, even -> A pair load is
// 8B-aligned and contiguous -> single b64 load per lane per step).
// Out-of-range rows/cols are CLAMPED (not zeroed): a clamped A row only
// corrupts the same (unstored) D row; a clamped B col only corrupts the same
// (unstored) D col. No branches in the K loop.
// ---------------------------------------------------------------------------
__global__ __launch_bounds__(32)
void k_gemm_wmma(const float* __restrict__ A, const float* __restrict__ B,
                 float* __restrict__ C, int M, int N, int K) {
    int lane = threadIdx.x;        // 0..31
    int half = lane >> 4;          // 0/1
    int l16  = lane & 15;
    int row0 = blockIdx.x * 16;
    int col0 = blockIdx.y * 16;

    int ar = min(row0 + l16, M - 1);
    int bc = min(col0 + l16, N - 1);

    // A lane layout: VGPR0 <-> K = k0+2*half, VGPR1 <-> K = k0+2*half+1 (contiguous)
    const float* Ap = A + (size_t)ar * K + half * 2;
    // B lane layout: VGPR0 <-> row K = k0+2*half, VGPR1 <-> row +1 (stride N)
    const float* Bp = B + (size_t)(half * 2) * N + bc;

    v8f acc = {};
    for (int k0 = 0; k0 < K; k0 += 4) {
        v2f a = *(const v2f*)Ap;   // one 8-byte load
        v2f b;
        b.x = Bp[0];
        b.y = Bp[N];
        acc = __builtin_amdgcn_wmma_f32_16x16x4_f32(false, a, false, b,
                                                    (short)0, acc, false, false);
        Ap += 4;
        Bp += (size_t)4 * N;
    }
    for (int r = 0; r < 8; ++r) {
        int rr = row0 + half * 8 + r;   // C: VGPR r, lanes 0-15 -> M=r, 16-31 -> M=r+8
        int cc = col0 + l16;
        if (rr < M && cc < N) C[(size_t)rr * N + cc] = acc[r];
    }
}

// ---------------------------------------------------------------------------
// xws prefix-add + mean over f:  xwmean[b,n] = mean_g( XWS0[b,g,n] + jc*sum_{f<g} XW[b,f,n] )
// ---------------------------------------------------------------------------
__global__ void k_xws(const float* __restrict__ XWS0, const float* __restrict__ XW,
                      const int* __restrict__ jumpp, float* __restrict__ xwmean) {
    int tid = blockIdx.x * blockDim.x + threadIdx.x;
    if (tid >= BB * NN) return;
    int b = tid / NN, n = tid % NN;
    float jc = (float)jumpp[0];
    float w0 = XW[(b * 3 + 0) * NN + n];
    float w1 = XW[(b * 3 + 1) * NN + n];
    float r0 = XWS0[(b * 3 + 0) * NN + n];
    float r1 = XWS0[(b * 3 + 1) * NN + n] + jc * w0;
    float r2 = XWS0[(b * 3 + 2) * NN + n] + jc * (w0 + w1);
    xwmean[b * NN + n] = (r0 + r1 + r2) * (1.f / 3.f);
}

// ---------------------------------------------------------------------------
// Per-node pooled weights: Wn(500x3x48), wws(500x8), wwt(500x8), biasN(500x64)
// ---------------------------------------------------------------------------
__global__ __launch_bounds__(224)
void k_node(const float* __restrict__ E, const float* __restrict__ wpool,
            const float* __restrict__ wsp, const float* __restrict__ wtp,
            const float* __restrict__ bpool, float* __restrict__ Wn,
            float* __restrict__ wws, float* __restrict__ wwt,
            float* __restrict__ biasN) {
    int n = blockIdx.x, t = threadIdx.x;    // t in [0,224)
    __shared__ float ei[EMB];
    if (t < EMB) ei[t] = E[n * EMB + t];
    __syncthreads();
    if (t < 144) {                      // weights_pool (10,3,1,48) -> d*144 + t
        float a = 0.f;
        for (int d = 0; d < EMB; ++d) a += ei[d] * wpool[d * 144 + t];
        Wn[n * 144 + t] = a;
    } else if (t < 152) {
        int oo = t - 144; float a = 0.f;
        for (int d = 0; d < EMB; ++d) a += ei[d] * wsp[d * 8 + oo];
        wws[n * 8 + oo] = a;
    } else if (t < 160) {
        int oo = t - 152; float a = 0.f;
        for (int d = 0; d < EMB; ++d) a += ei[d] * wtp[d * 8 + oo];
        wwt[n * 8 + oo] = a;
    } else {
        int o = t - 160; float a = 0.f;
        for (int d = 0; d < EMB; ++d) a += ei[d] * bpool[d * 64 + o];
        biasN[n * 64 + o] = a;
    }
}

// ---------------------------------------------------------------------------
// Fused conv(3x3, 12->8ch) + bias + relu + per-(b,h)-row partial mean/max.
// Block = (h, b); LDS stages 3 rows x 502 cols per t-plane.
// part[(b*500+h)*8 + oc]: oc<4 row-sums (mean conv), oc>=4 row-maxes (max conv).
// ---------------------------------------------------------------------------
__global__ __launch_bounds__(256)
void k_conv(const float* __restrict__ MPG, const float* __restrict__ wm,
            const float* __restrict__ bm, const float* __restrict__ wx,
            const float* __restrict__ bx, float* __restrict__ part) {
    int h = blockIdx.x, b = blockIdx.y;
    int tid = threadIdx.x;
    __shared__ float wl[864];           // [0,432): mean w, [432,864): max w
    __shared__ float buf[3][512];       // 3 rows, cols -1..500 at index c+1
    __shared__ float red[256];

    for (int i = tid; i < 432; i += 256) { wl[i] = wm[i]; wl[432 + i] = wx[i]; }

    float accm[2][4], accx[2][4];
    for (int p = 0; p < 2; ++p)
        for (int oc = 0; oc < 4; ++oc) { accm[p][oc] = bm[oc]; accx[p][oc] = bx[oc]; }

    for (int t = 0; t < TT; ++t) {
        const float* plane = MPG + (size_t)(b * TT + t) * (NN * NN);
        __syncthreads();
        for (int i = tid; i < 3 * 502; i += 256) {
            int dr = i / 502, c = i % 502;
            int rr = h + dr - 1, col = c - 1;
            float v = 0.f;
            if (rr >= 0 && rr < NN && col >= 0 && col < NN) {
                v = plane[rr * NN + col];
                if (t < TT - 1) __builtin_prefetch(&plane[(NN * NN) + rr * NN + col], 0, 1);
            }
            buf[dr][c] = v;
        }
        __syncthreads();
        for (int p = 0; p < 2; ++p) {
            int w = tid + p * 256;
            if (w < NN) {
                float v00 = buf[0][w], v01 = buf[0][w + 1], v02 = buf[0][w + 2];
                float v10 = buf[1][w], v11 = buf[1][w + 1], v12 = buf[1][w + 2];
                float v20 = buf[2][w], v21 = buf[2][w + 1], v22 = buf[2][w + 2];
                for (int oc = 0; oc < 4; ++oc) {
                    const float* W9 = &wl[oc * 108 + t * 9];
                    accm[p][oc] += v00 * W9[0] + v01 * W9[1] + v02 * W9[2]
                                 + v10 * W9[3] + v11 * W9[4] + v12 * W9[5]
                                 + v20 * W9[6] + v21 * W9[7] + v22 * W9[8];
                    const float* X9 = &wl[432 + oc * 108 + t * 9];
                    accx[p][oc] += v00 * X9[0] + v01 * X9[1] + v02 * X9[2]
                                 + v10 * X9[3] + v11 * X9[4] + v12 * X9[5]
                                 + v20 * X9[6] + v21 * X9[7] + v22 * X9[8];
                }
            }
        }
    }

    float lsum[4], lmax[4];
    for (int oc = 0; oc < 4; ++oc) {
        float a0 = (tid       < NN) ? fmaxf(accm[0][oc], 0.f) : 0.f;
        float a1 = (tid + 256 < NN) ? fmaxf(accm[1][oc], 0.f) : 0.f;
        lsum[oc] = a0 + a1;
        float m0 = (tid       < NN) ? fmaxf(accx[0][oc], 0.f) : 0.f;
        float m1 = (tid + 256 < NN) ? fmaxf(accx[1][oc], 0.f) : 0.f;
        lmax[oc] = fmaxf(m0, m1);   // relu output >= 0, so 0 is a safe identity
    }
    for (int oc = 0; oc < 8; ++oc) {
        bool ismax = oc >= 4;
        __syncthreads();
        red[tid] = ismax ? lmax[oc - 4] : lsum[oc];
        __syncthreads();
        for (int s = 128; s > 0; s >>= 1) {
            if (tid < s) red[tid] = ismax ? fmaxf(red[tid], red[tid + s])
                                          : red[tid] + red[tid + s];
            __syncthreads();
        }
        if (tid == 0) part[((size_t)b * NN + h) * 8 + oc] = red[0];
    }
}

__global__ __launch_bounds__(256)
void k_conv_reduce(const float* __restrict__ part, float* __restrict__ topo) {
    int b = blockIdx.x, tid = threadIdx.x;
    __shared__ float red[256];
    for (int oc = 0; oc < 8; ++oc) {
        bool ismax = oc >= 4;
        float acc = 0.f;
        for (int hh = tid; hh < NN; hh += 256) {
            float v = part[((size_t)b * NN + hh) * 8 + oc];
            acc = ismax ? fmaxf(acc, v) : acc + v;
        }
        red[tid] = acc;
        __syncthreads();
        for (int s = 128; s > 0; s >>= 1) {
            if (tid < s) red[tid] = ismax ? fmaxf(red[tid], red[tid + s])
                                          : red[tid] + red[tid + s];
            __syncthreads();
        }
        if (tid == 0) topo[b * 8 + oc] = ismax ? red[0] : red[0] * (1.0f / 250000.0f);
        __syncthreads();
    }
}

// ---------------------------------------------------------------------------
// Final assembly: out[b,n,o] = concat(x_sconv, x_twconv_s, x_wconv_t) + bias
// ---------------------------------------------------------------------------
__global__ __launch_bounds__(64)
void k_final(const float* __restrict__ x, const float* __restrict__ Y1,
             const float* __restrict__ Y2, const float* __restrict__ Wn,
             const float* __restrict__ wws, const float* __restrict__ wwt,
             const float* __restrict__ xwmean, const float* __restrict__ xt,
             const float* __restrict__ topo, const float* __restrict__ biasN,
             float* __restrict__ out) {
    int n = blockIdx.x, b = blockIdx.y, o = threadIdx.x;
    float val;
    if (o < 48) {
        float xs0 = x[b * NN + n];
        float xs1 = Y1[n * BB + b];
        float xs2 = Y2[n * BB + b];
        val = xs0 * Wn[n * 144 + o] + xs1 * Wn[n * 144 + 48 + o]
            + xs2 * Wn[n * 144 + 96 + o];
    } else if (o < 56) {
        int oo = o - 48;
        val = wws[n * 8 + oo] * xwmean[b * NN + n] * topo[b * 8 + oo];
    } else {
        int oo = o - 56;
        val = wwt[n * 8 + oo] * xt[b * NN + n];
    }
    out[((size_t)b * NN + n) * 64 + o] = val + biasN[n * 64 + o];
}

// ---------------------------------------------------------------------------
extern "C" void kernel_launch(void* const* d_in, const int* in_sizes, int n_in,
                              void* d_out, int out_size, void* d_ws, size_t ws_size,
                              hipStream_t stream) {
    const float* x      = (const float*)d_in[0];
    const float* xwin   = (const float*)d_in[1];
    const float* E      = (const float*)d_in[2];
    const float* adj    = (const float*)d_in[3];
    const float* MPG    = (const float*)d_in[4];
    const int*   bidx   = (const int*)d_in[5];
    const int*   fixedp = (const int*)d_in[6];
    const int*   stayp  = (const int*)d_in[7];
    const int*   jumpp  = (const int*)d_in[8];
    const float* wpool  = (const float*)d_in[9];
    const float* wsp    = (const float*)d_in[10];
    const float* wtp    = (const float*)d_in[11];
    const float* bpool  = (const float*)d_in[12];
    const float* Tp     = (const float*)d_in[13];
    const float* cmw    = (const float*)d_in[14];
    const float* cmb    = (const float*)d_in[15];
    const float* cxw    = (const float*)d_in[16];
    const float* cxb    = (const float*)d_in[17];
    float* out = (float*)d_out;

    float* ws = (float*)d_ws;
    float* S      = ws;             // 250000
    float* rowsum = S + 250000;     // 512
    float* Xt     = rowsum + 512;   // 16000  (500x32)
    float* Y1     = Xt + 16000;     // 16000
    float* Y2     = Y1 + 16000;     // 16000
    float* XW     = Y2 + 16000;     // 48000  (96x500)
    float* XWS0   = XW + 48000;     // 48000
    float* xwmean = XWS0 + 48000;   // 16000
    float* xt     = xwmean + 16000; // 16000
    float* Wn     = xt + 16000;     // 72000
    float* wws    = Wn + 72000;     // 4000
    float* wwt    = wws + 4000;     // 4000
    float* biasN  = wwt + 4000;     // 32000
    float* part   = biasN + 32000;  // 128000
    float* topo   = part + 128000;  // 256

    k_rowS<<<NN, 512, 0, stream>>>(E, adj, S, rowsum, fixedp, stayp);
    k_coldiv<<<(NN * NN + 255) / 256, 256, 0, stream>>>(S, rowsum);
    k_gather<<<(BB * NN + 255) / 256, 256, 0, stream>>>(x, xwin, Tp, bidx, Xt, XW, xt);

    // Y1 = S (500x500) * Xt (500x32); Y2 = S * Y1; XWS0 = XW (96x500) * S
    k_gemm_wmma<<<dim3(32, 2),  32, 0, stream>>>(S,  Xt, Y1,   NN, BB, NN);
    k_gemm_wmma<<<dim3(32, 2),  32, 0, stream>>>(S,  Y1, Y2,   NN, BB, NN);
    k_gemm_wmma<<<dim3(6, 32),  32, 0, stream>>>(XW, S,  XWS0, 96, NN, NN);

    k_xws<<<(BB * NN + 255) / 256, 256, 0, stream>>>(XWS0, XW, jumpp, xwmean);
    k_node<<<NN, 224, 0, stream>>>(E, wpool, wsp, wtp, bpool, Wn, wws, wwt, biasN);

    k_conv<<<dim3(NN, BB), 256, 0, stream>>>(MPG, cmw, cmb, cxw, cxb, part);
    k_conv_reduce<<<BB, 256, 0, stream>>>(part, topo);

    k_final<<<dim3(NN, BB), 64, 0, stream>>>(x, Y1, Y2, Wn, wws, wwt,
                                             xwmean, xt, topo, biasN, out);
}